// Model_14654428414079
// MI455X (gfx1250) — compile-verified
//
#include <hip/hip_runtime.h>
#include <hip/hip_bf16.h>

#define BB 128
#define SS 512
#define HH 768
#define CC 3

typedef __attribute__((ext_vector_type(2))) float v2f;
typedef __attribute__((ext_vector_type(8))) float v8f;

// ---------------------------------------------------------------------------
// Kernel 1: per-batch span mask + masked mean-pool of embedding rows.
// Only masked rows are read from HBM (mask is very sparse: ~1-10 rows/batch),
// cutting traffic ~50x vs. the reference's full 192 MiB read.
// ---------------------------------------------------------------------------
__global__ __launch_bounds__(256) void mask_mean_kernel(
    const float* __restrict__ emb,      // [B,S,H]
    const int*   __restrict__ offsets,  // [B,S,2]
    const int*   __restrict__ spans,    // [B,2]
    float*       __restrict__ feats) {  // [B,H] (workspace)
  const int b   = blockIdx.x;
  const int tid = threadIdx.x;

  __shared__ int s_match[SS];
  __shared__ int s_nonspec[SS];
  __shared__ int s_idx[SS];
  __shared__ int s_firstMatch, s_firstBrk, s_cnt;

  if (tid == 0) { s_firstMatch = SS; s_firstBrk = SS; s_cnt = 0; }
  __syncthreads();

  const int span0 = spans[b * 2 + 0];
  const int span1 = spans[b * 2 + 1];

  for (int s = tid; s < SS; s += blockDim.x) {
    const int os0 = offsets[((size_t)b * SS + s) * 2 + 0];
    const int os1 = offsets[((size_t)b * SS + s) * 2 + 1];
    const int nonspec = (os1 != 0);
    const int match   = nonspec && (os0 >= span0) && (os1 <= span1);
    s_match[s]   = match;
    s_nonspec[s] = nonspec;
    if (match) atomicMin(&s_firstMatch, s);
  }
  __syncthreads();

  const int fm = s_firstMatch;
  for (int s = tid; s < SS; s += blockDim.x) {
    if (s > fm && s_nonspec[s] && !s_match[s]) atomicMin(&s_firstBrk, s);
  }
  __syncthreads();

  const int fb = s_firstBrk;
  if (fm >= SS) {
    // no match anywhere -> mask[0] = 1 fallback (matches reference)
    if (tid == 0) { s_idx[0] = 0; s_cnt = 1; }
  } else {
    for (int s = tid; s < SS; s += blockDim.x) {
      if (s_match[s] && s < fb) {
        const int p = atomicAdd(&s_cnt, 1);
        s_idx[p] = s;  // order irrelevant for a sum
      }
    }
  }
  __syncthreads();

  const int   cnt = s_cnt;
  const float inv = 1.0f / (float)cnt;

  for (int h = tid; h < HH; h += blockDim.x) {
    float acc = 0.0f;
    for (int i = 0; i < cnt; ++i) {
      const int s = s_idx[i];
      if (i + 1 < cnt) {
        // gfx1250 global_prefetch_b8 hint for the next masked row
        __builtin_prefetch(&emb[((size_t)b * SS + s_idx[i + 1]) * HH + h], 0, 0);
      }
      acc += emb[((size_t)b * SS + s) * HH + h];
    }
    feats[(size_t)b * HH + h] = acc * inv;
  }
}

// ---------------------------------------------------------------------------
// Kernel 2: logits = feats(128x768) @ W(768x3) + bias via V_WMMA_F32_16X16X4_F32.
// 8 blocks (one 16-row tile each) x 4 waves; each wave accumulates a K/4
// slice (48 WMMAs, short dependency chain), partials reduced through LDS.
// B-operand is built branch-free: clamped column address + 0/1 multiplier,
// so EXEC stays all-1s and no saveexec predication appears in the hot loop.
// A layout (ISA 16x4 f32): lanes 0-15 hold M=lane, K={k,k+1}; lanes 16-31
// hold M=lane-16, K={k+2,k+3}  -> one contiguous b64 load per lane per step.
// ---------------------------------------------------------------------------
__global__ __launch_bounds__(128) void logits_wmma_kernel(
    const float* __restrict__ feats,  // [B,H]
    const float* __restrict__ W,      // [H,C]
    const float* __restrict__ bias,   // [C]
    float*       __restrict__ out) {  // [B,C]
  const int  tile = blockIdx.x;       // 8 tiles of 16 batch rows
  const int  tid  = threadIdx.x;
  const int  lane = tid & 31;
  const int  wave = tid >> 5;         // 4 waves, each owns a K slice
  const int  nlo  = lane & 15;
  const bool hiH  = lane >= 16;
  const int  k0   = hiH ? 2 : 0;

  // Branch-free B-column handling: valid address always, dead columns x0.
  const int   nloc = (nlo < CC) ? nlo : 0;
  const float cm   = (nlo < CC) ? 1.0f : 0.0f;

  const int    m    = tile * 16 + nlo;
  const float* arow = feats + (size_t)m * HH;

  const int kbeg = wave * (HH / 4);   // 192 floats of K per wave
  const int kend = kbeg + (HH / 4);

  v8f c = {};
  for (int k = kbeg; k < kend; k += 4) {
    const v2f a = *(const v2f*)(arow + k + k0);  // contiguous {K=k0, K=k0+1}
    v2f bb;
    bb.x = W[(k + k0 + 0) * CC + nloc] * cm;
    bb.y = W[(k + k0 + 1) * CC + nloc] * cm;
    // 8 args: (neg_a, A, neg_b, B, c_mod, C, reuse_a, reuse_b)
    c = __builtin_amdgcn_wmma_f32_16x16x4_f32(
        false, a, false, bb, (short)0, c, false, false);
  }

  // Cross-wave reduction of the 16x16 partial C tiles through LDS.
  __shared__ float red[4 * 256];
#pragma unroll
  for (int v = 0; v < 8; ++v) red[wave * 256 + lane * 8 + v] = c[v];
  __syncthreads();

  if (wave == 0) {
    // C/D layout: VGPR v -> M = v + (lane>=16 ? 8 : 0), N = lane&15
    if (nlo < CC) {
      const float bv = bias[nlo];
#pragma unroll
      for (int v = 0; v < 8; ++v) {
        const int idx = lane * 8 + v;
        const float s =
            red[idx] + red[256 + idx] + red[512 + idx] + red[768 + idx];
        const int row = tile * 16 + v + (hiH ? 8 : 0);
        out[row * CC + nlo] = s + bv;
      }
    }
  }
}

// ---------------------------------------------------------------------------
// Launch
// ---------------------------------------------------------------------------
extern "C" void kernel_launch(void* const* d_in, const int* in_sizes, int n_in,
                              void* d_out, int out_size, void* d_ws, size_t ws_size,
                              hipStream_t stream) {
  (void)in_sizes; (void)n_in; (void)out_size; (void)ws_size;

  const float* emb     = (const float*)d_in[0];  // [128,512,768] f32
  const float* W       = (const float*)d_in[1];  // [768,3] f32
  const float* bias    = (const float*)d_in[2];  // [3] f32
  const int*   offsets = (const int*)d_in[3];    // [128,512,2] i32
  const int*   spans   = (const int*)d_in[4];    // [128,2] i32

  float* feats = (float*)d_ws;                   // [128,768] f32 scratch
  float* out   = (float*)d_out;                  // [128,3] f32

  mask_mean_kernel<<<BB, 256, 0, stream>>>(emb, offsets, spans, feats);
  logits_wmma_kernel<<<BB / 16, 128, 0, stream>>>(feats, W, bias, out);
}